// Partitioner_51651276702441
// MI455X (gfx1250) — compile-verified
//
#include <hip/hip_runtime.h>
#include <hip/hip_bf16.h>
#include <math.h>

#define NB   32      // batch
#define NN   1024    // nodes
#define ND   128     // feature dim
#define NH   128     // hidden dim
#define NC   16      // max clusters
#define NEGV (-1e9f)

typedef __attribute__((ext_vector_type(2))) float v2f;
typedef __attribute__((ext_vector_type(8))) float v8f;

// ---------------------------------------------------------------------------
// WMMA GEMM: XW[m, j] = sum_k X[m,k] * W[k,j],  m in [0,32768), k,j in [0,128)
// W = Ws1 rows 0..127 (the x-part of the seed MLP first layer).
// One block = 8 waves; block stages W (64KB) in LDS; each wave computes a
// 16-row strip against all 8 column tiles (8 accumulators), K in steps of 4.
// ---------------------------------------------------------------------------
__global__ __launch_bounds__(256) void k_gemm_xw(const float* __restrict__ X,
                                                 const float* __restrict__ W,
                                                 float* __restrict__ XW) {
    __shared__ float Wl[NH * NH];             // 64 KB
    for (int i = threadIdx.x; i < NH * NH; i += 256) Wl[i] = W[i];
    __syncthreads();

    const int wave = threadIdx.x >> 5;        // wave32
    const int lane = threadIdx.x & 31;
    const int half = lane >> 4;               // 0: K+0/K+1   1: K+2/K+3
    const int l16  = lane & 15;
    const int rowBase = (blockIdx.x * 8 + wave) * 16;

    v8f acc[8];
    const v8f vzero = {0.f, 0.f, 0.f, 0.f, 0.f, 0.f, 0.f, 0.f};
#pragma unroll
    for (int ct = 0; ct < 8; ++ct) acc[ct] = vzero;

    const float* xrow = X + (size_t)(rowBase + l16) * ND;
    for (int kk = 0; kk < ND; kk += 4) {
        const int k0 = kk + 2 * half;
        v2f a;
        a.x = xrow[k0 + 0];
        a.y = xrow[k0 + 1];
#pragma unroll
        for (int ct = 0; ct < 8; ++ct) {
            const int n = ct * 16 + l16;
            v2f bf;
            bf.x = Wl[(k0 + 0) * NH + n];
            bf.y = Wl[(k0 + 1) * NH + n];
            acc[ct] = __builtin_amdgcn_wmma_f32_16x16x4_f32(
                false, a, false, bf, (short)0, acc[ct], false, false);
        }
    }
#pragma unroll
    for (int ct = 0; ct < 8; ++ct) {
        const int col = ct * 16 + l16;
#pragma unroll
        for (int r = 0; r < 8; ++r) {
            const int row = rowBase + r + 8 * half;
            XW[(size_t)row * NH + col] = acc[ct][r];
        }
    }
}

// ---------------------------------------------------------------------------
// Init helpers
// ---------------------------------------------------------------------------
__global__ __launch_bounds__(256) void k_zero(float* a, int n) {
    int i = blockIdx.x * 256 + threadIdx.x;
    if (i < n) a[i] = 0.0f;
}

__global__ __launch_bounds__(256) void k_cadj(float* cadj) {
    int i = blockIdx.x * 256 + threadIdx.x;   // NB*NC*NC = 8192
    int r = (i >> 4) & 15, c = i & 15;
    cadj[i] = (r == c) ? 0.0f : 1.0f;
}

__global__ __launch_bounds__(256) void k_initb(const unsigned char* __restrict__ mask,
                                               int* avail, float* totalNodes,
                                               float* assignedTotal) {
    int b = blockIdx.x;
    __shared__ float cnt[256];
    float c = 0.f;
    for (int n = threadIdx.x; n < NN; n += 256) {
        int m = mask[b * NN + n] ? 1 : 0;
        avail[b * NN + n] = m;
        c += (float)m;
    }
    cnt[threadIdx.x] = c; __syncthreads();
    for (int s = 128; s > 0; s >>= 1) {
        if (threadIdx.x < s) cnt[threadIdx.x] += cnt[threadIdx.x + s];
        __syncthreads();
    }
    if (threadIdx.x == 0) { totalNodes[b] = cnt[0]; assignedTotal[b] = 0.f; }
}

__global__ __launch_bounds__(128) void k_xmean(const float* __restrict__ x,
                                               const unsigned char* __restrict__ mask,
                                               float* xmean) {
    int b = blockIdx.x, k = threadIdx.x;
    float s = 0.f;
    for (int n = 0; n < NN; ++n) {
        float m = mask[b * NN + n] ? 1.0f : 0.0f;
        s += m * x[((size_t)b * NN + n) * ND + k];
    }
    xmean[b * ND + k] = s * (1.0f / (float)NN);
}

__global__ __launch_bounds__(128) void k_gctx0(const float* __restrict__ xmean,
                                               const float* __restrict__ Wc,
                                               const float* __restrict__ bc, float* gctx) {
    int b = blockIdx.x, j = threadIdx.x;
    __shared__ float xs[ND];
    xs[j] = xmean[b * ND + j];
    __syncthreads();
    float a = bc[j];
    for (int k = 0; k < ND; ++k) a += xs[k] * Wc[k * NH + j];
    gctx[b * NH + j] = a;
}

// ---------------------------------------------------------------------------
// Per-iteration kernels
// ---------------------------------------------------------------------------
// 1) active flags, anyActive, t[b,:] = gctx @ Ws1[128:,:] + bs1   (single block)
__global__ __launch_bounds__(1024) void k_prep(const float* __restrict__ gctx,
                                               const float* __restrict__ Ws1,
                                               const float* __restrict__ bs1,
                                               const float* __restrict__ assignedTotal,
                                               const float* __restrict__ totalNodes,
                                               int* active, int* anyActive,
                                               float* t, int c) {
    __shared__ int s_any;
    if (threadIdx.x == 0) s_any = 0;
    __syncthreads();
    if (threadIdx.x < NB) {
        int b = threadIdx.x, a;
        if (c == 0) a = 1;
        else {
            float tot = totalNodes[b];
            float cov = assignedTotal[b] / (tot + 1e-8f);
            a = (cov < 0.95f) && (tot > 0.0f);
        }
        active[b] = a;
        if (a) atomicOr(&s_any, 1);
    }
    __syncthreads();
    if (threadIdx.x == 0) *anyActive = s_any;
    for (int i = threadIdx.x; i < NB * NH; i += blockDim.x) {
        int b = i >> 7, j = i & 127;
        float acc = bs1[j];
        for (int k = 0; k < NH; ++k) acc += gctx[b * NH + k] * Ws1[(ND + k) * NH + j];
        t[i] = acc;
    }
}

// 2) seed_logits[b,n] = relu(XW[b,n,:] + t[b,:]) . Ws2 + bs2
__global__ __launch_bounds__(256) void k_logits(const float* __restrict__ XW,
                                                const float* __restrict__ t,
                                                const float* __restrict__ Ws2,
                                                const float* __restrict__ bs2,
                                                float* seed_logits) {
    __shared__ float ts[NH], w2[NH];
    int b = blockIdx.x;
    if (threadIdx.x < NH) {
        ts[threadIdx.x] = t[b * NH + threadIdx.x];
        w2[threadIdx.x] = Ws2[threadIdx.x];
    }
    __syncthreads();
    int n = blockIdx.y * 256 + threadIdx.x;
    const float* row = XW + ((size_t)b * NN + n) * NH;
    float acc = bs2[0];
#pragma unroll 8
    for (int k = 0; k < NH; ++k) acc += fmaxf(row[k] + ts[k], 0.0f) * w2[k];
    seed_logits[b * NN + n] = acc;
}

// 3) gumbel argmax seed selection; write seed into assignment, update avail
__global__ __launch_bounds__(256) void k_seed(const float* __restrict__ seed_logits,
                                              const float* __restrict__ noise,
                                              const int* __restrict__ active,
                                              int* avail, int* idxOut, int* validOut,
                                              float* assignOut, int c) {
    int b = blockIdx.x;
    __shared__ float sv[256];
    __shared__ int   si[256];
    __shared__ int   s_any;
    if (threadIdx.x == 0) s_any = 0;
    __syncthreads();
    float bv = -INFINITY; int bi = 0x7fffffff; int anySel = 0;
    int act = active[b];
    for (int n = threadIdx.x; n < NN; n += 256) {
        int sel = avail[b * NN + n] && act;
        anySel |= sel;
        float lg = sel ? seed_logits[b * NN + n] : NEGV;
        float u  = noise[((size_t)c * NB + b) * NN + n];
        float g  = -logf(-logf(u + 1e-8f) + 1e-8f);
        float v  = lg + g;                   // TAU==1, argmax of softmax == argmax of this
        if (v > bv || (v == bv && n < bi)) { bv = v; bi = n; }
    }
    if (anySel) atomicOr(&s_any, 1);
    sv[threadIdx.x] = bv; si[threadIdx.x] = bi;
    __syncthreads();
    for (int s = 128; s > 0; s >>= 1) {
        if (threadIdx.x < s) {
            float ov = sv[threadIdx.x + s]; int oi = si[threadIdx.x + s];
            if (ov > sv[threadIdx.x] || (ov == sv[threadIdx.x] && oi < si[threadIdx.x])) {
                sv[threadIdx.x] = ov; si[threadIdx.x] = oi;
            }
        }
        __syncthreads();
    }
    if (threadIdx.x == 0) {
        int idx = si[0];
        int valid = s_any;                   // == sel_mask.any() && active
        idxOut[b] = idx; validOut[b] = valid;
        if (valid) {
            assignOut[((size_t)b * NN + idx) * NC + c] = 1.0f;
            avail[b * NN + idx] = 0;
        }
    }
}

// 4) 2-hop BFS from seed (sparse frontier; only needed adjacency rows touched)
__global__ __launch_bounds__(256) void k_bfs(const float* __restrict__ adj,
                                             const unsigned char* __restrict__ mask,
                                             const int* __restrict__ idxIn,
                                             const int* __restrict__ active,
                                             const int* __restrict__ avail, int* cands) {
    int b = blockIdx.x;
    __shared__ unsigned char reach[NN];
    __shared__ int f1[NN];
    __shared__ int f1n;
    int seed = idxIn[b];
    for (int n = threadIdx.x; n < NN; n += 256) reach[n] = (n == seed);
    if (threadIdx.x == 0) f1n = 0;
    __syncthreads();
    // hop 1: neighbors of seed
    for (int n = threadIdx.x; n < NN; n += 256) {
        if (!reach[n] && mask[b * NN + n] &&
            adj[((size_t)b * NN + seed) * NN + n] > 0.0f) {
            int p = atomicAdd(&f1n, 1);
            f1[p] = n;
            reach[n] = 1;
        }
    }
    __syncthreads();
    // hop 2: neighbors of hop-1 frontier
    int cnt = f1n;
    for (int m = threadIdx.x; m < NN; m += 256) {
        if (!reach[m] && mask[b * NN + m]) {
            for (int i = 0; i < cnt; ++i) {
                if (adj[((size_t)b * NN + f1[i]) * NN + m] > 0.0f) { reach[m] = 1; break; }
            }
        }
    }
    __syncthreads();
    int act = active[b];
    for (int m = threadIdx.x; m < NN; m += 256)
        cands[b * NN + m] = avail[b * NN + m] && reach[m] && act;
}

// 5) size-prediction MLP -> add_needed
__global__ __launch_bounds__(128) void k_size(const float* __restrict__ x,
                                              const float* __restrict__ gctx,
                                              const int* __restrict__ idxIn,
                                              const float* __restrict__ Wp1,
                                              const float* __restrict__ bp1,
                                              const float* __restrict__ Wp2,
                                              const float* __restrict__ bp2,
                                              int* addNeeded) {
    int b = blockIdx.x, j = threadIdx.x;
    __shared__ float in[2 * ND];
    __shared__ float hid[NH];
    int seed = idxIn[b];
    in[j]      = x[((size_t)b * NN + seed) * ND + j];
    in[ND + j] = gctx[b * NH + j];
    __syncthreads();
    float acc = bp1[j];
    for (int k = 0; k < 2 * ND; ++k) acc += in[k] * Wp1[k * NH + j];
    hid[j] = fmaxf(acc, 0.0f);
    __syncthreads();
    if (j == 0) {
        float lg[3];
        for (int o = 0; o < 3; ++o) {
            float a = bp2[o];
            for (int k = 0; k < NH; ++k) a += hid[k] * Wp2[k * 3 + o];
            lg[o] = a;                       // TAU == 1
        }
        float mx = fmaxf(lg[0], fmaxf(lg[1], lg[2]));
        float e0 = expf(lg[0] - mx), e1 = expf(lg[1] - mx), e2 = expf(lg[2] - mx);
        float s = e0 + e1 + e2;
        float pred = (e0 * 1.0f + e1 * 2.0f + e2 * 3.0f) / s;
        addNeeded[b] = (int)floorf(fmaxf(pred - 1.0f, 0.0f));
    }
}

// 6) top-2 candidates, select extras, update assignment/avail, member list
__global__ __launch_bounds__(256) void k_topk(const float* __restrict__ seed_logits,
                                              const int* __restrict__ cands,
                                              const int* __restrict__ addNeeded,
                                              const int* __restrict__ validIn,
                                              const int* __restrict__ idxIn,
                                              int* avail, float* assignOut,
                                              float* assignedTotal,
                                              int* members, int* mcount, int c) {
    int b = blockIdx.x;
    __shared__ float sv[256];
    __shared__ int   si[256];
    __shared__ int   top[2];
    for (int r = 0; r < 2; ++r) {
        float bv = -INFINITY; int bi = 0x7fffffff;
        for (int n = threadIdx.x; n < NN; n += 256) {
            if (r == 1 && n == top[0]) continue;
            float v = cands[b * NN + n] ? seed_logits[b * NN + n] : -INFINITY;
            if (v > bv || (v == bv && n < bi)) { bv = v; bi = n; }
        }
        sv[threadIdx.x] = bv; si[threadIdx.x] = bi;
        __syncthreads();
        for (int s = 128; s > 0; s >>= 1) {
            if (threadIdx.x < s) {
                float ov = sv[threadIdx.x + s]; int oi = si[threadIdx.x + s];
                if (ov > sv[threadIdx.x] || (ov == sv[threadIdx.x] && oi < si[threadIdx.x])) {
                    sv[threadIdx.x] = ov; si[threadIdx.x] = oi;
                }
            }
            __syncthreads();
        }
        if (threadIdx.x == 0) top[r] = si[0];
        __syncthreads();
    }
    if (threadIdx.x == 0) {
        int valid = validIn[b];
        int an = addNeeded[b];
        int mc = 0;
        if (valid) members[b * 3 + mc++] = idxIn[b];
        for (int r = 0; r < 2; ++r) {
            int ok = cands[b * NN + top[r]];
            if ((r < an) && ok && valid) {
                assignOut[((size_t)b * NN + top[r]) * NC + c] = 1.0f;
                avail[b * NN + top[r]] = 0;
                members[b * 3 + mc++] = top[r];
            }
        }
        mcount[b] = mc;
        assignedTotal[b] += (float)mc;
    }
}

// 7) cluster embedding -> feats output; gctx += 0.1*(emb@Wc + bc) if any active
__global__ __launch_bounds__(128) void k_emb(const float* __restrict__ x,
                                             const int* __restrict__ members,
                                             const int* __restrict__ mcount,
                                             const int* __restrict__ active,
                                             const int* __restrict__ anyActive,
                                             const float* __restrict__ Wc,
                                             const float* __restrict__ bc,
                                             float* gctx, float* featsOut, int c) {
    int b = blockIdx.x, j = threadIdx.x;
    __shared__ float emb[ND];
    int mc = mcount[b];
    float s = 0.0f;
    for (int i = 0; i < mc; ++i)
        s += x[((size_t)b * NN + members[b * 3 + i]) * ND + j];
    float csize = fmaxf((float)mc, 1.0f);
    float e = (s / csize) * (active[b] ? 1.0f : 0.0f);
    emb[j] = e;
    featsOut[((size_t)b * NC + c) * ND + j] = e;
    __syncthreads();
    if (*anyActive) {
        float acc = bc[j];
        for (int k = 0; k < ND; ++k) acc += emb[k] * Wc[k * NH + j];
        gctx[b * NH + j] += 0.1f * acc;
    }
}

// ---------------------------------------------------------------------------
extern "C" void kernel_launch(void* const* d_in, const int* in_sizes, int n_in,
                              void* d_out, int out_size, void* d_ws, size_t ws_size,
                              hipStream_t stream) {
    const float* x    = (const float*)d_in[0];
    const float* adj  = (const float*)d_in[1];
    const unsigned char* mask = (const unsigned char*)d_in[2];
    const float* noise= (const float*)d_in[3];
    const float* Wc   = (const float*)d_in[4];
    const float* bc   = (const float*)d_in[5];
    const float* Ws1  = (const float*)d_in[6];
    const float* bs1  = (const float*)d_in[7];
    const float* Ws2  = (const float*)d_in[8];
    const float* bs2  = (const float*)d_in[9];
    const float* Wp1  = (const float*)d_in[10];
    const float* bp1  = (const float*)d_in[11];
    const float* Wp2  = (const float*)d_in[12];
    const float* bp2  = (const float*)d_in[13];

    float* out_feats  = (float*)d_out;                       // [B,NC,D]
    float* out_cadj   = out_feats + NB * NC * ND;            // [B,NC,NC]
    float* out_assign = out_cadj + NB * NC * NC;             // [B,N,NC]

    char* ws = (char*)d_ws;
    size_t off = 0;
    auto take = [&](size_t bytes) -> char* {
        char* p = ws + off;
        off += (bytes + 255) & ~(size_t)255;
        return p;
    };
    float* XW          = (float*)take(sizeof(float) * NB * NN * NH);   // 16 MB
    float* gctx        = (float*)take(sizeof(float) * NB * NH);
    float* t           = (float*)take(sizeof(float) * NB * NH);
    float* xmean       = (float*)take(sizeof(float) * NB * ND);
    float* seed_logits = (float*)take(sizeof(float) * NB * NN);
    float* totalNodes  = (float*)take(sizeof(float) * NB);
    float* assignedTot = (float*)take(sizeof(float) * NB);
    int*   active      = (int*)take(sizeof(int) * NB);
    int*   anyActive   = (int*)take(sizeof(int));
    int*   idx         = (int*)take(sizeof(int) * NB);
    int*   valid       = (int*)take(sizeof(int) * NB);
    int*   addNeeded   = (int*)take(sizeof(int) * NB);
    int*   mcount      = (int*)take(sizeof(int) * NB);
    int*   members     = (int*)take(sizeof(int) * NB * 3);
    int*   avail       = (int*)take(sizeof(int) * NB * NN);
    int*   cands       = (int*)take(sizeof(int) * NB * NN);

    // ---- init ----
    k_zero<<<(NB * NN * NC + 255) / 256, 256, 0, stream>>>(out_assign, NB * NN * NC);
    k_cadj<<<(NB * NC * NC + 255) / 256, 256, 0, stream>>>(out_cadj);
    k_initb<<<NB, 256, 0, stream>>>(mask, avail, totalNodes, assignedTot);
    k_xmean<<<NB, 128, 0, stream>>>(x, mask, xmean);
    k_gctx0<<<NB, 128, 0, stream>>>(xmean, Wc, bc, gctx);

    // ---- big iteration-invariant GEMM with WMMA: XW = x @ Ws1[:128,:] ----
    k_gemm_xw<<<(NB * NN) / 128, 256, 0, stream>>>(x, Ws1, XW);

    // ---- 16 sequential cluster rounds ----
    for (int c = 0; c < NC; ++c) {
        k_prep<<<1, 1024, 0, stream>>>(gctx, Ws1, bs1, assignedTot, totalNodes,
                                       active, anyActive, t, c);
        k_logits<<<dim3(NB, NN / 256), 256, 0, stream>>>(XW, t, Ws2, bs2, seed_logits);
        k_seed<<<NB, 256, 0, stream>>>(seed_logits, noise, active, avail,
                                       idx, valid, out_assign, c);
        k_bfs<<<NB, 256, 0, stream>>>(adj, mask, idx, active, avail, cands);
        k_size<<<NB, 128, 0, stream>>>(x, gctx, idx, Wp1, bp1, Wp2, bp2, addNeeded);
        k_topk<<<NB, 256, 0, stream>>>(seed_logits, cands, addNeeded, valid, idx,
                                       avail, out_assign, assignedTot,
                                       members, mcount, c);
        k_emb<<<NB, 128, 0, stream>>>(x, members, mcount, active, anyActive,
                                      Wc, bc, gctx, out_feats, c);
    }
}